// VectorwiseSparsity_9105330668183
// MI455X (gfx1250) — compile-verified
//
#include <hip/hip_runtime.h>

#define B_ 64
#define T_ 1024
#define C_ 1024
#define K_ 64
#define TT 128   // t-rows per scatter block

typedef __attribute__((ext_vector_type(2))) float v2f;
typedef __attribute__((ext_vector_type(8))) float v8f;

// ---------------------------------------------------------------------------
// Kernel 1: scores[b,t] = dot(x[b,t,:], W) + bias using V_WMMA_F32_16X16X4_F32.
// One wave computes a 16-row tile of scores. A = 16x4 tile of x (f32 layout:
// lanes 0-15 hold K=0,1 in 2 VGPRs; lanes 16-31 hold K=2,3). B = 4x16 with
// W[k] broadcast across all N columns, so D[m][n] == score(m) for every n.
// D layout: VGPR r -> M=r (lanes 0-15) / M=8+r (lanes 16-31); lane 0 and
// lane 16 together hold a full column, so they write the 16 scores.
// ---------------------------------------------------------------------------
__global__ __launch_bounds__(128) void scores_wmma(const float* __restrict__ x,
                                                   const float* __restrict__ W,
                                                   const float* __restrict__ bias_p,
                                                   float* __restrict__ scores) {
  const int wave = blockIdx.x * (blockDim.x >> 5) + (threadIdx.x >> 5);
  const int lane = threadIdx.x & 31;
  const int tilesPerB = T_ / 16;
  const int b  = wave / tilesPerB;
  const int t0 = (wave % tilesPerB) * 16;
  const int row  = t0 + (lane & 15);
  const int koff = (lane >> 4) * 2;   // lanes 0-15: K {0,1}; lanes 16-31: K {2,3}

  const float* __restrict__ xrow = x + ((size_t)b * T_ + row) * C_ + koff;
  const float* __restrict__ wp   = W + koff;

  v8f acc = {};
#pragma unroll 4
  for (int c = 0; c < C_; c += 4) {
    v2f a, bm;
    a.x  = xrow[c];     a.y  = xrow[c + 1];   // contiguous float2 per lane
    bm.x = wp[c];       bm.y = wp[c + 1];     // W broadcast over N columns
    acc = __builtin_amdgcn_wmma_f32_16x16x4_f32(false, a, false, bm,
                                                (short)0, acc, false, false);
  }

  const float bias = bias_p[0];
  if (lane == 0) {
#pragma unroll
    for (int r = 0; r < 8; ++r) scores[(size_t)b * T_ + t0 + r] = acc[r] + bias;
  } else if (lane == 16) {
#pragma unroll
    for (int r = 0; r < 8; ++r) scores[(size_t)b * T_ + t0 + 8 + r] = acc[r] + bias;
  }
}

// ---------------------------------------------------------------------------
// Kernel 2: top-K by rank counting. Keys are (monotonic(f32) << 32) | (~t),
// so keys are unique, ranks are a permutation, and ties break toward the
// smaller index (matching lax.top_k). Each thread counts keys greater than
// its own via uniform LDS reads (hardware broadcast), then the K winners
// write their time index compacted by rank.
// ---------------------------------------------------------------------------
__global__ __launch_bounds__(1024) void topk_rank(const float* __restrict__ scores,
                                                  int* __restrict__ sel) {
  __shared__ unsigned long long keys[T_];
  const int b = blockIdx.x;
  const int t = threadIdx.x;

  unsigned int u = __float_as_uint(scores[(size_t)b * T_ + t]);
  u ^= ((unsigned int)((int)u >> 31)) | 0x80000000u;          // order-preserving map
  const unsigned long long key =
      ((unsigned long long)u << 32) | (unsigned int)(T_ - 1 - t);
  keys[t] = key;
  __syncthreads();

  int rank = 0;
  for (int i = 0; i < T_; ++i) rank += (keys[i] > key) ? 1 : 0;
  if (rank < K_) sel[b * K_ + rank] = t;
}

// ---------------------------------------------------------------------------
// Kernel 3: out[b,t,c] = (c in sel[b]) ? x[b,c,t] : 0.
// One block per (b, 128-row t-tile). Stage the 64 selected rows' t-slice in
// padded LDS (row pitch 129 -> bank = (k + tt) % 64, conflict-free), build a
// column->k map, then write the whole tile with coalesced float4 stores.
// Output is written exactly once (no memset pass).
// ---------------------------------------------------------------------------
__global__ __launch_bounds__(256) void scatter_out(const float* __restrict__ x,
                                                   const int* __restrict__ sel,
                                                   float* __restrict__ out) {
  __shared__ float gbuf[K_][TT + 1];
  __shared__ int   colk[C_];
  __shared__ int   selk[K_];

  const int tilesPerB = T_ / TT;
  const int b   = blockIdx.x / tilesPerB;
  const int t0  = (blockIdx.x % tilesPerB) * TT;
  const int tid = threadIdx.x;

  if (tid < K_) selk[tid] = sel[b * K_ + tid];
  for (int c = tid; c < C_; c += 256) colk[c] = -1;
  __syncthreads();

  if (tid < K_) colk[selk[tid]] = tid;

  // Gather the selected rows' slice: coalesced 128-float runs along t.
  for (int i = tid; i < K_ * TT; i += 256) {
    const int k  = i / TT;
    const int tt = i % TT;
    gbuf[k][tt] = x[((size_t)b * T_ + selk[k]) * C_ + (t0 + tt)];
  }
  __syncthreads();

  // Emit the tile: 32 lanes cover 512 contiguous bytes per float4 store.
  for (int i = tid; i < TT * (C_ / 4); i += 256) {
    const int tt = i / (C_ / 4);
    const int c  = (i % (C_ / 4)) * 4;
    const int k0 = colk[c + 0], k1 = colk[c + 1];
    const int k2 = colk[c + 2], k3 = colk[c + 3];
    float4 v;
    v.x = (k0 >= 0) ? gbuf[k0][tt] : 0.0f;
    v.y = (k1 >= 0) ? gbuf[k1][tt] : 0.0f;
    v.z = (k2 >= 0) ? gbuf[k2][tt] : 0.0f;
    v.w = (k3 >= 0) ? gbuf[k3][tt] : 0.0f;
    *(float4*)(out + ((size_t)b * T_ + (t0 + tt)) * C_ + c) = v;
  }
}

extern "C" void kernel_launch(void* const* d_in, const int* in_sizes, int n_in,
                              void* d_out, int out_size, void* d_ws, size_t ws_size,
                              hipStream_t stream) {
  const float* x    = (const float*)d_in[0];   // [B, T, C] f32
  const float* W    = (const float*)d_in[1];   // [C, 1]    f32
  const float* bias = (const float*)d_in[2];   // [1]       f32
  float* out = (float*)d_out;                  // [B, T, C] f32

  float* scores = (float*)d_ws;                                // B*T floats
  int*   sel    = (int*)((char*)d_ws + (size_t)B_ * T_ * sizeof(float)); // B*K ints

  // 1 wave per 16-score tile: B*T/16 = 4096 waves -> 1024 blocks of 128.
  scores_wmma<<<(B_ * (T_ / 16)) / 4, 128, 0, stream>>>(x, W, bias, scores);
  topk_rank  <<<B_, 1024, 0, stream>>>(scores, sel);
  scatter_out<<<B_ * (T_ / TT), 256, 0, stream>>>(x, sel, out);
}